// DeepseekV3MoE_4501125726823
// MI455X (gfx1250) — compile-verified
//
#include <hip/hip_runtime.h>

// ---------------------------------------------------------------------------
// DeepseekV3 MoE block for MI455X (gfx1250), fp32 end-to-end using
// V_WMMA_F32_16X16X4_F32. Memory-bound (~1 GB HBM traffic @ 23.3 TB/s vs
// ~1 TFLOP of matrix math), so fp32 WMMA costs nothing vs bf16 and keeps full
// fidelity. Staging uses GLOBAL_LOAD_ASYNC_TO_LDS_B128 (ASYNCcnt) with
// double-buffered LDS so weight/activation streams stay in flight under the
// WMMA inner loop.
// ---------------------------------------------------------------------------

typedef __attribute__((ext_vector_type(2))) float v2f;
typedef __attribute__((ext_vector_type(4))) float v4f;
typedef __attribute__((ext_vector_type(8))) float v8f;
typedef int i32x4 __attribute__((vector_size(16)));   // matches async-LDS builtin param

// Problem constants (fixed by the reference).
constexpr int HIDDEN    = 2048;
constexpr int INTER     = 8192;
constexpr int MOE_INTER = 1024;
constexpr int NEXP      = 8;
constexpr int TOKENS    = 4 * 2048;   // B * S

// GEMM tiling: 256 threads = 8 wave32s laid out 2 (M) x 4 (N).
// Block tile 64x128, wave tile 32x32 = 2x2 WMMA 16x16 f32 accumulators.
constexpr int TM   = 64;              // token rows per block
constexpr int TN   = 128;             // output cols per block
constexpr int KC   = 32;              // K chunk staged in LDS
constexpr int LDST = KC + 4;          // 36 floats: rows 16B-aligned, bank-conflict-free

__device__ __forceinline__ v8f wmma_f32(v2f a, v2f b, v8f c) {
  // D = A(16x4) * B(4x16) + C(16x16), fp32.
  return __builtin_amdgcn_wmma_f32_16x16x4_f32(
      /*neg_a=*/false, a, /*neg_b=*/false, b,
      /*c_mod=*/(short)0, c, /*reuse_a=*/false, /*reuse_b=*/false);
}

// --- gfx1250 async global->LDS staging (ASYNCcnt) with safe fallback -------
#if __has_builtin(__builtin_amdgcn_global_load_async_to_lds_b128)
#define ASYNC_LDS 1
#else
#define ASYNC_LDS 0
#endif

__device__ __forceinline__ void stage16(const float* __restrict__ g,
                                        float* __restrict__ l) {
#if ASYNC_LDS
  __builtin_amdgcn_global_load_async_to_lds_b128(
      (__attribute__((address_space(1))) i32x4*)const_cast<float*>(g),
      (__attribute__((address_space(3))) i32x4*)l, /*offset=*/0, /*cpol=*/0);
#else
  *(v4f*)l = *(const v4f*)g;   // sync fallback: VGPR round trip
#endif
}

__device__ __forceinline__ void stage_wait() {
#if ASYNC_LDS
#if __has_builtin(__builtin_amdgcn_s_wait_asynccnt)
  __builtin_amdgcn_s_wait_asynccnt(0);
#else
  asm volatile("s_wait_asynccnt 0x0" ::: "memory");
#endif
#endif
}

// ---------------------------------------------------------------------------
// Router: logits = x @ Wr^T (E=8), top-2, softmax over the pair, mask
// (idx < NUM_LOCAL), sum -> comb_w[t]. One wave per token.
// ---------------------------------------------------------------------------
__global__ __launch_bounds__(256) void moe_router_kernel(
    const float* __restrict__ X, const float* __restrict__ Wr,
    float* __restrict__ comb) {
  const int lane = threadIdx.x & 31;
  const int wave = threadIdx.x >> 5;
  const int t = blockIdx.x * 8 + wave;
  const float* xr = X + (size_t)t * HIDDEN;

  float acc[NEXP];
#pragma unroll
  for (int e = 0; e < NEXP; ++e) acc[e] = 0.f;

  for (int h = lane; h < HIDDEN; h += 32) {
    const float xv = xr[h];
#pragma unroll
    for (int e = 0; e < NEXP; ++e) acc[e] += xv * Wr[e * HIDDEN + h];
  }
#pragma unroll
  for (int e = 0; e < NEXP; ++e) {
#pragma unroll
    for (int off = 16; off > 0; off >>= 1)
      acc[e] += __shfl_xor(acc[e], off, 32);
  }
  if (lane == 0) {
    int i0 = 0; float v0 = acc[0];
#pragma unroll
    for (int e = 1; e < NEXP; ++e) if (acc[e] > v0) { v0 = acc[e]; i0 = e; }
    int i1 = -1; float v1 = -3.4e38f;
#pragma unroll
    for (int e = 0; e < NEXP; ++e)
      if (e != i0 && acc[e] > v1) { v1 = acc[e]; i1 = e; }
    // softmax over (v0, v1), v0 >= v1
    const float e1 = __expf(v1 - v0);
    const float s  = 1.f + e1;
    const float w0 = 1.f / s, w1 = e1 / s;
    const float cw = (i0 < NEXP ? w0 : 0.f) + (i1 < NEXP ? w1 : 0.f);
    comb[t] = cw;
  }
}

// ---------------------------------------------------------------------------
// Fused gate/up projection + SiLU(g)*u.  act[t,n] = silu(x@Wg^T) * (x@Wu^T)
// Wg/Wu are (N, H) row-major -> K (=H) contiguous for both operands.
// ---------------------------------------------------------------------------
__global__ __launch_bounds__(256) void moe_gu_kernel(
    const float* __restrict__ X,    // [TOKENS, HIDDEN]
    const float* __restrict__ Wg,   // [N, HIDDEN]
    const float* __restrict__ Wu,   // [N, HIDDEN]
    float* __restrict__ act,        // [TOKENS, N]
    int N) {
  __shared__ float Xs[2][TM * LDST];
  __shared__ float Gs[2][TN * LDST];
  __shared__ float Us[2][TN * LDST];

  const int tid  = threadIdx.x;
  const int lane = tid & 31;
  const int wave = tid >> 5;
  const int wm   = wave >> 2;          // 0..1 (M dir)
  const int wn   = wave & 3;           // 0..3 (N dir)
  const int lr   = lane & 15;          // row/col within 16
  const int lk   = (lane >> 4) << 1;   // K sub-offset: 0 or 2

  const int m0 = blockIdx.x * TM;
  const int n0 = blockIdx.y * TN;

  const v8f zero = {0.f, 0.f, 0.f, 0.f, 0.f, 0.f, 0.f, 0.f};
  v8f accG[2][2], accU[2][2];
#pragma unroll
  for (int i = 0; i < 2; ++i)
#pragma unroll
    for (int j = 0; j < 2; ++j) { accG[i][j] = zero; accU[i][j] = zero; }

  const int ldr = tid >> 3;            // 0..31
  const int ldc = (tid & 7) << 2;      // 0..28 step 4

  auto stage = [&](int k0, int buf) {
#pragma unroll
    for (int rr = 0; rr < TM; rr += 32)
      stage16(X + (size_t)(m0 + ldr + rr) * HIDDEN + k0 + ldc,
              &Xs[buf][(ldr + rr) * LDST + ldc]);
#pragma unroll
    for (int rr = 0; rr < TN; rr += 32) {
      stage16(Wg + (size_t)(n0 + ldr + rr) * HIDDEN + k0 + ldc,
              &Gs[buf][(ldr + rr) * LDST + ldc]);
      stage16(Wu + (size_t)(n0 + ldr + rr) * HIDDEN + k0 + ldc,
              &Us[buf][(ldr + rr) * LDST + ldc]);
    }
  };

  constexpr int NK = HIDDEN / KC;
  stage(0, 0);
  stage_wait();
  __syncthreads();

  for (int ki = 0; ki < NK; ++ki) {
    const int cur = ki & 1;
    if (ki + 1 < NK) stage((ki + 1) * KC, cur ^ 1);  // prefetch next chunk

#pragma unroll
    for (int kk = 0; kk < KC; kk += 4) {
      v2f a[2], bg[2], bu[2];
#pragma unroll
      for (int mi = 0; mi < 2; ++mi)
        a[mi] = *(const v2f*)&Xs[cur][(wm * 32 + mi * 16 + lr) * LDST + kk + lk];
#pragma unroll
      for (int ni = 0; ni < 2; ++ni) {
        bg[ni] = *(const v2f*)&Gs[cur][(wn * 32 + ni * 16 + lr) * LDST + kk + lk];
        bu[ni] = *(const v2f*)&Us[cur][(wn * 32 + ni * 16 + lr) * LDST + kk + lk];
      }
#pragma unroll
      for (int mi = 0; mi < 2; ++mi)
#pragma unroll
        for (int ni = 0; ni < 2; ++ni) {
          accG[mi][ni] = wmma_f32(a[mi], bg[ni], accG[mi][ni]);
          accU[mi][ni] = wmma_f32(a[mi], bu[ni], accU[mi][ni]);
        }
    }

    if (ki + 1 < NK) {
      stage_wait();        // my async loads for chunk ki+1 are in LDS
      __syncthreads();     // everyone done reading cur & done staging nxt
    }
  }

  // C/D layout: VGPR v -> row v + 8*(lane>=16), col = lane & 15.
  const int hi8 = (lane >> 4) * 8;
#pragma unroll
  for (int mi = 0; mi < 2; ++mi)
#pragma unroll
    for (int ni = 0; ni < 2; ++ni) {
      const v8f g = accG[mi][ni];
      const v8f u = accU[mi][ni];
      const int col = n0 + wn * 32 + ni * 16 + lr;
#pragma unroll
      for (int v = 0; v < 8; ++v) {
        const float gv = g[v];
        const float sig = 1.f / (1.f + __expf(-gv));
        const float a = gv * sig * u[v];
        const int row = m0 + wm * 32 + mi * 16 + hi8 + v;
        act[(size_t)row * N + col] = a;
      }
    }
}

// ---------------------------------------------------------------------------
// Down projection: out[t,h] (+)= scale(t) * sum_k act[t,k] * Wd[h,k]
// Wd is (HIDDEN, K) row-major. comb==nullptr -> no scale.
// ---------------------------------------------------------------------------
__global__ __launch_bounds__(256) void moe_down_kernel(
    const float* __restrict__ A,     // [TOKENS, K]
    const float* __restrict__ Wd,    // [HIDDEN, K]
    const float* __restrict__ comb,  // [TOKENS] or nullptr
    float* __restrict__ out,         // [TOKENS, HIDDEN]
    int K, int accumulate) {
  __shared__ float As[2][TM * LDST];
  __shared__ float Bs[2][TN * LDST];

  const int tid  = threadIdx.x;
  const int lane = tid & 31;
  const int wave = tid >> 5;
  const int wm   = wave >> 2;
  const int wn   = wave & 3;
  const int lr   = lane & 15;
  const int lk   = (lane >> 4) << 1;

  const int m0 = blockIdx.x * TM;
  const int n0 = blockIdx.y * TN;

  const v8f zero = {0.f, 0.f, 0.f, 0.f, 0.f, 0.f, 0.f, 0.f};
  v8f acc[2][2];
#pragma unroll
  for (int i = 0; i < 2; ++i)
#pragma unroll
    for (int j = 0; j < 2; ++j) acc[i][j] = zero;

  const int ldr = tid >> 3;
  const int ldc = (tid & 7) << 2;

  auto stage = [&](int k0, int buf) {
#pragma unroll
    for (int rr = 0; rr < TM; rr += 32)
      stage16(A + (size_t)(m0 + ldr + rr) * K + k0 + ldc,
              &As[buf][(ldr + rr) * LDST + ldc]);
#pragma unroll
    for (int rr = 0; rr < TN; rr += 32)
      stage16(Wd + (size_t)(n0 + ldr + rr) * K + k0 + ldc,
              &Bs[buf][(ldr + rr) * LDST + ldc]);
  };

  const int NK = K / KC;
  stage(0, 0);
  stage_wait();
  __syncthreads();

  for (int ki = 0; ki < NK; ++ki) {
    const int cur = ki & 1;
    if (ki + 1 < NK) stage((ki + 1) * KC, cur ^ 1);

#pragma unroll
    for (int kk = 0; kk < KC; kk += 4) {
      v2f a[2], b[2];
#pragma unroll
      for (int mi = 0; mi < 2; ++mi)
        a[mi] = *(const v2f*)&As[cur][(wm * 32 + mi * 16 + lr) * LDST + kk + lk];
#pragma unroll
      for (int ni = 0; ni < 2; ++ni)
        b[ni] = *(const v2f*)&Bs[cur][(wn * 32 + ni * 16 + lr) * LDST + kk + lk];
#pragma unroll
      for (int mi = 0; mi < 2; ++mi)
#pragma unroll
        for (int ni = 0; ni < 2; ++ni)
          acc[mi][ni] = wmma_f32(a[mi], b[ni], acc[mi][ni]);
    }

    if (ki + 1 < NK) {
      stage_wait();
      __syncthreads();
    }
  }

  const int hi8 = (lane >> 4) * 8;
#pragma unroll
  for (int mi = 0; mi < 2; ++mi) {
#pragma unroll
    for (int v = 0; v < 8; ++v) {
      const int row = m0 + wm * 32 + mi * 16 + hi8 + v;
      const float cw = comb ? comb[row] : 1.f;
#pragma unroll
      for (int ni = 0; ni < 2; ++ni) {
        const int col = n0 + wn * 32 + ni * 16 + lr;
        const size_t o = (size_t)row * HIDDEN + col;
        const float val = acc[mi][ni][v] * cw;
        out[o] = accumulate ? (out[o] + val) : val;
      }
    }
  }
}

// ---------------------------------------------------------------------------
// Launch: router -> routed gate/up -> routed down (writes out, scaled by
// comb_w) -> shared gate/up -> shared down (accumulates into out).
// Workspace: comb_w (32KB) | act_routed (32MB) | act_shared (256MB).
// ---------------------------------------------------------------------------
extern "C" void kernel_launch(void* const* d_in, const int* in_sizes, int n_in,
                              void* d_out, int out_size, void* d_ws, size_t ws_size,
                              hipStream_t stream) {
  (void)in_sizes; (void)n_in; (void)out_size; (void)ws_size;

  const float* X   = (const float*)d_in[0];  // hidden_states [T, H]
  const float* Wr  = (const float*)d_in[1];  // router_weight [8, H]
  const float* shG = (const float*)d_in[2];  // [INTER, H]
  const float* shU = (const float*)d_in[3];  // [INTER, H]
  const float* shD = (const float*)d_in[4];  // [H, INTER]
  const float* rG  = (const float*)d_in[5];  // [MOE_INTER, H]
  const float* rU  = (const float*)d_in[6];  // [MOE_INTER, H]
  const float* rD  = (const float*)d_in[7];  // [H, MOE_INTER]
  float* out = (float*)d_out;

  char* ws = (char*)d_ws;
  float* comb   = (float*)ws;                                   // 8192 f32
  float* act_r  = (float*)(ws + (64 << 10));                    // [T, MOE_INTER]
  float* act_sh = (float*)(ws + (64 << 10) +
                           (size_t)TOKENS * MOE_INTER * sizeof(float));

  const dim3 blk(256);

  moe_router_kernel<<<TOKENS / 8, blk, 0, stream>>>(X, Wr, comb);

  moe_gu_kernel<<<dim3(TOKENS / TM, MOE_INTER / TN), blk, 0, stream>>>(
      X, rG, rU, act_r, MOE_INTER);
  moe_down_kernel<<<dim3(TOKENS / TM, HIDDEN / TN), blk, 0, stream>>>(
      act_r, rD, comb, out, MOE_INTER, /*accumulate=*/0);

  moe_gu_kernel<<<dim3(TOKENS / TM, INTER / TN), blk, 0, stream>>>(
      X, shG, shU, act_sh, INTER);
  moe_down_kernel<<<dim3(TOKENS / TM, HIDDEN / TN), blk, 0, stream>>>(
      act_sh, shD, /*comb=*/nullptr, out, INTER, /*accumulate=*/1);
}